// RowColAttention_19215683682328
// MI455X (gfx1250) — compile-verified
//
#include <hip/hip_runtime.h>
#include <hip/hip_bf16.h>

typedef float v2f __attribute__((ext_vector_type(2)));
typedef float v8f __attribute__((ext_vector_type(8)));

static __device__ __forceinline__ v8f wmma_f32_4(v2f a, v2f b, v8f c) {
  // D = A(16x4) * B(4x16) + C(16x16), fp32, emits v_wmma_f32_16x16x4_f32
  return __builtin_amdgcn_wmma_f32_16x16x4_f32(false, a, false, b, (short)0, c,
                                               false, false);
}

#define ZERO8 {0.f,0.f,0.f,0.f,0.f,0.f,0.f,0.f}

// ---------------------------------------------------------------------------
// GEMM: C[M,N] = A[M,K] * B[K,N] (+ bias[N]); row-major everywhere.
// Block = 256 threads = 8 waves; block tile 128(M) x 64(N); K chunked by 64.
// B chunk staged in LDS transposed (stride 66 f32) so each WMMA B-fragment is
// a single aligned ds_load_b64. A fragments are direct global b64 loads.
// ---------------------------------------------------------------------------
__global__ __launch_bounds__(256) void gemm128x64_wmma(
    const float* __restrict__ A, const float* __restrict__ B,
    const float* __restrict__ bias, float* __restrict__ C,
    int M, int N, int K) {
  __shared__ float bt[64 * 66];  // BT[n][k], n,k in [0,64), pad to 66

  const int m0 = blockIdx.x * 128;
  const int n0 = blockIdx.y * 64;
  const int tid = threadIdx.x;
  const int w = tid >> 5;
  const int lane = tid & 31;
  const int L = lane & 15;     // column / row-in-tile index
  const int hi = lane >> 4;    // selects K pair {0,1} vs {2,3}
  const int arow = m0 + w * 16 + L;

  v8f acc[4] = {ZERO8, ZERO8, ZERO8, ZERO8};

  const int c4 = (tid & 15) * 4;  // 4 consecutive B columns per thread
  const int r0 = tid >> 4;        // 16 rows per pass

  for (int kc = 0; kc < K; kc += 64) {
    // ---- stage B[kc:kc+64, n0:n0+64] into LDS transposed ----
#pragma unroll
    for (int p = 0; p < 4; ++p) {
      const int r = r0 + p * 16;
      const float4 bv =
          *(const float4*)&B[(size_t)(kc + r) * N + n0 + c4];
      bt[(c4 + 0) * 66 + r] = bv.x;
      bt[(c4 + 1) * 66 + r] = bv.y;
      bt[(c4 + 2) * 66 + r] = bv.z;
      bt[(c4 + 3) * 66 + r] = bv.w;
    }
    __syncthreads();
    // ---- 16 K-steps of 4 ----
#pragma unroll
    for (int kk = 0; kk < 64; kk += 4) {
      const v2f av = *(const v2f*)&A[(size_t)arow * K + kc + kk + hi * 2];
#pragma unroll
      for (int t = 0; t < 4; ++t) {
        const v2f bf = *(const v2f*)&bt[(t * 16 + L) * 66 + kk + hi * 2];
        acc[t] = wmma_f32_4(av, bf, acc[t]);
      }
    }
    __syncthreads();
  }

  // ---- epilogue: optional bias, write C ----
  const int crow = m0 + w * 16 + hi * 8;
#pragma unroll
  for (int t = 0; t < 4; ++t) {
    const int col = n0 + t * 16 + L;
    const float bs = bias ? bias[col] : 0.0f;
#pragma unroll
    for (int r = 0; r < 8; ++r)
      C[(size_t)(crow + r) * N + col] = acc[t][r] + bs;
  }
}

// ---------------------------------------------------------------------------
// Attention: one block per (s,b,h). Scores kept in LDS (256 x 258 f32 padded)
// plus 256 reciprocal row sums -> ~265 KB dynamic LDS (CDNA5 WGP has 320 KB).
// Phase 1: sim = scale * q k^T, masked, via WMMA -> LDS
// Phase 2: row softmax (thread t owns row t; stride 258 is bank-conflict-free)
// Phase 3: out = P V via WMMA (P A-frags are ds_load_b64 from LDS)
// qkv layout: [s][b][n][768] with q = h*32, k = 256+h*32, v = 512+h*32
// out layout: [s][b][n][256], column offset h*32
// ---------------------------------------------------------------------------
__global__ __launch_bounds__(256) void attn_kernel(
    const float* __restrict__ qkv, const int* __restrict__ mask,
    float* __restrict__ out) {
  extern __shared__ float lds[];
  float* simb = lds;               // [256][258]
  float* rinv = lds + 256 * 258;   // [256]

  const int bid = blockIdx.x;
  const int s = bid >> 5;
  const int b = (bid >> 3) & 3;
  const int h = bid & 7;

  const float* base = qkv + (size_t)((s * 4 + b) * 256) * 768;
  const float* qb = base + h * 32;
  const float* kb = base + 256 + h * 32;
  const float* vb = base + 512 + h * 32;
  const int* mk = mask + (size_t)s * 256 * 256;
  float* ob = out + (size_t)((s * 4 + b) * 256) * 256 + h * 32;

  const int tid = threadIdx.x;
  const int w = tid >> 5;
  const int lane = tid & 31;
  const int L = lane & 15;
  const int hi = lane >> 4;
  const float scale = 0.17677669529663687f;  // 32^-0.5
  const float NEG = -1.0e10f;

  // ---- phase 1: scores ----
  v2f qa[2][8];  // A-fragments for this wave's two 16-row q blocks
#pragma unroll
  for (int rb = 0; rb < 2; ++rb) {
    const int rbase = rb * 128 + w * 16;
#pragma unroll
    for (int c = 0; c < 8; ++c)
      qa[rb][c] = *(const v2f*)&qb[(size_t)(rbase + L) * 768 + c * 4 + hi * 2];
  }

  for (int j0 = 0; j0 < 256; j0 += 16) {
    v2f kf[8];  // B = k^T fragments: B[d][j] = k[j][d] (row-major k => b64)
#pragma unroll
    for (int c = 0; c < 8; ++c)
      kf[c] = *(const v2f*)&kb[(size_t)(j0 + L) * 768 + c * 4 + hi * 2];
#pragma unroll
    for (int rb = 0; rb < 2; ++rb) {
      v8f acc = ZERO8;
#pragma unroll
      for (int c = 0; c < 8; ++c) acc = wmma_f32_4(qa[rb][c], kf[c], acc);
      const int ib = rb * 128 + w * 16 + hi * 8;
#pragma unroll
      for (int r = 0; r < 8; ++r) {
        float sv = acc[r] * scale;
        sv = (mk[(ib + r) * 256 + j0 + L] == 0) ? NEG : sv;
        simb[(ib + r) * 258 + j0 + L] = sv;
      }
    }
  }
  __syncthreads();

  // ---- phase 2: softmax, one row per thread ----
  {
    float* row = simb + tid * 258;
    float m = -3.4e38f;
    for (int j = 0; j < 256; ++j) m = fmaxf(m, row[j]);
    float sum = 0.0f;
    for (int j = 0; j < 256; ++j) {
      const float e = __expf(row[j] - m);
      row[j] = e;
      sum += e;
    }
    rinv[tid] = 1.0f / sum;
  }
  __syncthreads();

  // ---- phase 3: out = P @ V ----
  v8f oacc[2][2] = {{ZERO8, ZERO8}, {ZERO8, ZERO8}};  // [rowblock][dtile]
  for (int jj = 0; jj < 256; jj += 4) {
    v2f va[2];  // B = V fragments: V[jj+2hi+{0,1}][d0+L]
#pragma unroll
    for (int dt = 0; dt < 2; ++dt) {
      v2f t;
      t.x = vb[(size_t)(jj + hi * 2) * 768 + dt * 16 + L];
      t.y = vb[(size_t)(jj + hi * 2 + 1) * 768 + dt * 16 + L];
      va[dt] = t;
    }
#pragma unroll
    for (int rb = 0; rb < 2; ++rb) {
      const v2f pa =
          *(const v2f*)&simb[(rb * 128 + w * 16 + L) * 258 + jj + hi * 2];
      oacc[rb][0] = wmma_f32_4(pa, va[0], oacc[rb][0]);
      oacc[rb][1] = wmma_f32_4(pa, va[1], oacc[rb][1]);
    }
  }
#pragma unroll
  for (int rb = 0; rb < 2; ++rb) {
    const int ib = rb * 128 + w * 16 + hi * 8;
#pragma unroll
    for (int r = 0; r < 8; ++r) {
      const float ri = rinv[ib + r];
      ob[(size_t)(ib + r) * 256 + L] = oacc[rb][0][r] * ri;
      ob[(size_t)(ib + r) * 256 + 16 + L] = oacc[rb][1][r] * ri;
    }
  }
}

// ---------------------------------------------------------------------------
extern "C" void kernel_launch(void* const* d_in, const int* in_sizes, int n_in,
                              void* d_out, int out_size, void* d_ws,
                              size_t ws_size, hipStream_t stream) {
  (void)in_sizes; (void)n_in; (void)out_size; (void)ws_size;
  const float* x    = (const float*)d_in[0];  // [64,4,256,256]
  const int*   mask = (const int*)d_in[1];    // [64,256,256]
  const float* Wqkv = (const float*)d_in[2];  // [256,768]
  const float* Wout = (const float*)d_in[3];  // [256,256]
  const float* bout = (const float*)d_in[4];  // [256]
  float* outp = (float*)d_out;                // [64,4,256,256]

  float* qkv = (float*)d_ws;                  // 65536*768 f32 (192 MB)
  float* ao  = qkv + (size_t)65536 * 768;     // 65536*256 f32 (64 MB)

  const dim3 blk(256);

  // qkv = x @ W_qkv
  gemm128x64_wmma<<<dim3(512, 12), blk, 0, stream>>>(x, Wqkv, nullptr, qkv,
                                                     65536, 768, 256);

  // per-head masked softmax attention
  const size_t smem = (size_t)(256 * 258 + 256) * sizeof(float);  // 265216 B
  hipFuncSetAttribute((const void*)attn_kernel,
                      hipFuncAttributeMaxDynamicSharedMemorySize, (int)smem);
  attn_kernel<<<dim3(2048), blk, smem, stream>>>(qkv, mask, ao);

  // out = ao @ W_out + b_out
  gemm128x64_wmma<<<dim3(512, 4), blk, 0, stream>>>(ao, Wout, bout, outp,
                                                    65536, 256, 256);
}